// GGSNN_40295383171461
// MI455X (gfx1250) — compile-verified
//
#include <hip/hip_runtime.h>

typedef float v2f __attribute__((ext_vector_type(2)));
typedef float v8f __attribute__((ext_vector_type(8)));

#define NGRAPH 256

// ---------------------------------------------------------------- init
__global__ void k_init(float* deg, float* agg1, float* agg2,
                       unsigned* pooled, int* cnt, int n) {
  long i = (long)blockIdx.x * blockDim.x + threadIdx.x;
  long total = (long)n * 32;
  if (i >= total) return;
  agg2[i] = 0.0f;
  if (i < (long)n * 3) agg1[i] = 0.0f;
  if (i < n) deg[i] = 1.0f;                 // self-loop contributes 1 to deg[row]
  if (i < NGRAPH * 64) pooled[i] = 0u;      // bits of +0.0f (post-ReLU vals >= 0)
  if (i < NGRAPH) cnt[i] = 0;
}

// ---------------------------------------------------------------- degree
__global__ void k_deg(const int* __restrict__ row, float* deg, int E) {
  int e = blockIdx.x * blockDim.x + threadIdx.x;
  if (e < E) atomicAdd(&deg[row[e]], 1.0f);
}

// ------------------------------------------- rsqrt + self-loop + counts
__global__ void k_norm(const float* __restrict__ pos, const int* __restrict__ batch,
                       float* deg_dis, float* agg1, int* cnt, int n) {
  int i = blockIdx.x * blockDim.x + threadIdx.x;
  if (i >= n) return;
  float d = deg_dis[i];                 // >= 1 always (self loop)
  float s = rsqrtf(d);
  deg_dis[i] = s;                       // becomes "dis"
  float s2 = s * s;                     // self-loop norm = dis[i]*dis[i]
  agg1[i*3+0] = s2 * pos[i*3+0];
  agg1[i*3+1] = s2 * pos[i*3+1];
  agg1[i*3+2] = s2 * pos[i*3+2];
  atomicAdd(&cnt[batch[i]], 1);
}

// ---------------------------------------------------------------- layer1 scatter
__global__ void k_edge1(const int* __restrict__ row, const int* __restrict__ col,
                        const float* __restrict__ dis, const float* __restrict__ pos,
                        float* agg1, int E) {
  int e = blockIdx.x * blockDim.x + threadIdx.x;
  if (e >= E) return;
  int r = row[e], c = col[e];
  float nrm = dis[r] * dis[c];
  atomicAdd(&agg1[c*3+0], nrm * pos[r*3+0]);
  atomicAdd(&agg1[c*3+1], nrm * pos[r*3+1]);
  atomicAdd(&agg1[c*3+2], nrm * pos[r*3+2]);
}

// ---------------------- layer1 MLP + ReLU, seed agg2 with layer2 self-loop
__global__ void k_mlp1(const float* __restrict__ agg1, const float* __restrict__ dis,
                       const float* __restrict__ W1, const float* __restrict__ b1,
                       float* h1, float* agg2, int n) {
  __shared__ float sW[96];
  __shared__ float sb[32];
  int t = threadIdx.x;
  if (t < 96) sW[t] = W1[t];
  if (t < 32) sb[t] = b1[t];
  __syncthreads();
  int i = blockIdx.x * blockDim.x + t;
  if (i >= n) return;
  float a0 = agg1[i*3+0], a1 = agg1[i*3+1], a2 = agg1[i*3+2];
  float s = dis[i];
  float s2 = s * s;
  #pragma unroll
  for (int j = 0; j < 32; ++j) {
    float h = fmaf(a0, sW[j*3+0], fmaf(a1, sW[j*3+1], fmaf(a2, sW[j*3+2], sb[j])));
    h = fmaxf(h, 0.0f);
    h1[i*32+j]   = h;
    agg2[i*32+j] = s2 * h;   // self-loop term of layer-2 aggregation
  }
}

// ----------------- layer2 scatter: one wave per edge, one lane per feature
__global__ void k_edge2(const int* __restrict__ row, const int* __restrict__ col,
                        const float* __restrict__ dis, const float* __restrict__ h1,
                        float* agg2, int E) {
  long gid = (long)blockIdx.x * blockDim.x + threadIdx.x;
  int e = (int)(gid >> 5);
  int j = (int)(gid & 31);
  if (e >= E) return;
  int r = row[e], c = col[e];
  float nrm = dis[r] * dis[c];
  atomicAdd(&agg2[(long)c*32 + j], nrm * h1[(long)r*32 + j]);
}

// ------- layer2 GEMM via V_WMMA_F32_16X16X4_F32 + fused bias/ReLU/max-pool
// Wave computes D[16,64] = A[16,32] x B[32,64]; A = agg2 rows, B = W2^T.
// A frag (16x4 f32): lane<16 -> (K=kb,kb+1), lane>=16 -> (K=kb+2,kb+3), M=lane%16
// B frag (4x16 f32): same K split, N=lane%16.  D: vgpr v -> M = v + 8*(lane/16).
__global__ void __launch_bounds__(256)
k_gemm_pool(const float* __restrict__ agg2, const float* __restrict__ W2,
            const float* __restrict__ bias2, const int* __restrict__ batch,
            unsigned* pooled, int nblk, int n) {
  int wave = (blockIdx.x * blockDim.x + threadIdx.x) >> 5;   // wave-uniform
  int lane = threadIdx.x & 31;
  if (wave >= nblk) return;          // uniform per wave -> EXEC stays all-1s
  int half = lane >> 4;
  int ln   = lane & 15;
  int rowBase = wave * 16;
  int mA = rowBase + ln;
  if (mA > n - 1) mA = n - 1;        // clamp (n is a multiple of 16 anyway)
  v8f acc0 = {}, acc1 = {}, acc2 = {}, acc3 = {};
  #pragma unroll
  for (int kt = 0; kt < 8; ++kt) {
    int kb = kt * 4 + half * 2;
    v2f a  = *(const v2f*)&agg2[(long)mA * 32 + kb];
    v2f b0 = *(const v2f*)&W2[(0*16 + ln) * 32 + kb];   // B[k][n] = W2[n*32+k]
    v2f b1 = *(const v2f*)&W2[(1*16 + ln) * 32 + kb];
    v2f b2 = *(const v2f*)&W2[(2*16 + ln) * 32 + kb];
    v2f b3 = *(const v2f*)&W2[(3*16 + ln) * 32 + kb];
    acc0 = __builtin_amdgcn_wmma_f32_16x16x4_f32(false, a, false, b0, (short)0, acc0, false, false);
    acc1 = __builtin_amdgcn_wmma_f32_16x16x4_f32(false, a, false, b1, (short)0, acc1, false, false);
    acc2 = __builtin_amdgcn_wmma_f32_16x16x4_f32(false, a, false, b2, (short)0, acc2, false, false);
    acc3 = __builtin_amdgcn_wmma_f32_16x16x4_f32(false, a, false, b3, (short)0, acc3, false, false);
  }
  float bb0 = bias2[ 0 + ln];
  float bb1 = bias2[16 + ln];
  float bb2 = bias2[32 + ln];
  float bb3 = bias2[48 + ln];
  #pragma unroll
  for (int v = 0; v < 8; ++v) {
    int m = rowBase + v + half * 8;
    if (m >= n) continue;
    int g = batch[m];
    unsigned* pg = pooled + g * 64;
    atomicMax(&pg[ 0 + ln], __float_as_uint(fmaxf(acc0[v] + bb0, 0.0f)));
    atomicMax(&pg[16 + ln], __float_as_uint(fmaxf(acc1[v] + bb1, 0.0f)));
    atomicMax(&pg[32 + ln], __float_as_uint(fmaxf(acc2[v] + bb2, 0.0f)));
    atomicMax(&pg[48 + ln], __float_as_uint(fmaxf(acc3[v] + bb3, 0.0f)));
  }
}

// ---------------------------------------------------------------- classifier
__global__ void k_out(const unsigned* __restrict__ pooled, const int* __restrict__ cnt,
                      const float* __restrict__ Wc, const float* __restrict__ bc,
                      float* out) {
  int t = blockIdx.x * blockDim.x + threadIdx.x;
  if (t >= NGRAPH * 2) return;
  int g = t >> 1, c = t & 1;
  bool empty = (cnt[g] == 0);
  float s = 0.0f;
  for (int j = 0; j < 64; ++j) {
    float v = empty ? -__builtin_inff() : __uint_as_float(pooled[g*64 + j]);
    s = fmaf(v, Wc[c*64 + j], s);
  }
  out[t] = s + bc[c];
}

extern "C" void kernel_launch(void* const* d_in, const int* in_sizes, int n_in,
                              void* d_out, int out_size, void* d_ws, size_t ws_size,
                              hipStream_t stream) {
  const float* pos  = (const float*)d_in[0];
  const int*   eidx = (const int*)  d_in[1];
  const int*   batch= (const int*)  d_in[2];
  const float* W1   = (const float*)d_in[3];
  const float* b1   = (const float*)d_in[4];
  const float* W2   = (const float*)d_in[5];
  const float* b2   = (const float*)d_in[6];
  const float* Wc   = (const float*)d_in[7];
  const float* bc   = (const float*)d_in[8];
  (void)n_in; (void)out_size; (void)ws_size;

  const int n = in_sizes[0] / 3;       // 100000
  const int E = in_sizes[1] / 2;       // 3200000
  const int* row = eidx;               // edge_index[0]
  const int* col = eidx + E;           // edge_index[1]

  size_t off = 0;
  auto bump = [&](size_t bytes) {
    size_t o = off;
    off += (bytes + 255) & ~(size_t)255;
    return o;
  };
  char* ws = (char*)d_ws;
  float*    deg    = (float*)   (ws + bump((size_t)n * 4));        // deg -> dis
  float*    agg1   = (float*)   (ws + bump((size_t)n * 3 * 4));
  float*    h1     = (float*)   (ws + bump((size_t)n * 32 * 4));
  float*    agg2   = (float*)   (ws + bump((size_t)n * 32 * 4));
  unsigned* pooled = (unsigned*)(ws + bump((size_t)NGRAPH * 64 * 4));
  int*      cnt    = (int*)     (ws + bump((size_t)NGRAPH * 4));

  const int TPB = 256;
  long initN = (long)n * 32;
  k_init <<<(initN + TPB - 1) / TPB, TPB, 0, stream>>>(deg, agg1, agg2, pooled, cnt, n);
  k_deg  <<<(E + TPB - 1) / TPB,     TPB, 0, stream>>>(row, deg, E);
  k_norm <<<(n + TPB - 1) / TPB,     TPB, 0, stream>>>(pos, batch, deg, agg1, cnt, n);
  k_edge1<<<(E + TPB - 1) / TPB,     TPB, 0, stream>>>(row, col, deg, pos, agg1, E);
  k_mlp1 <<<(n + TPB - 1) / TPB,     TPB, 0, stream>>>(agg1, deg, W1, b1, h1, agg2, n);
  long e2N = (long)E * 32;
  k_edge2<<<(e2N + TPB - 1) / TPB,   TPB, 0, stream>>>(row, col, deg, h1, agg2, E);
  int nblk = (n + 15) / 16;
  long gW = (long)nblk * 32;
  k_gemm_pool<<<(gW + TPB - 1) / TPB, TPB, 0, stream>>>(agg2, W2, b2, batch, pooled, nblk, n);
  k_out  <<<(NGRAPH * 2 + TPB - 1) / TPB, TPB, 0, stream>>>(pooled, cnt, Wc, bc, (float*)d_out);
}